// DGRec_24000277250316
// MI455X (gfx1250) — compile-verified
//
#include <hip/hip_runtime.h>
#include <hip/hip_bf16.h>
#include <math.h>

typedef __attribute__((ext_vector_type(2))) float v2f;
typedef __attribute__((ext_vector_type(8))) float v8f;
typedef __attribute__((ext_vector_type(4))) unsigned int u32x4;
typedef __attribute__((ext_vector_type(4))) int i32x4;
typedef __attribute__((ext_vector_type(8))) int i32x8;

#define E_DIM   100
#define L_DIM   20
#define B_DIM   100
#define S2_DIM  5
#define NGATE   400
#define V_DIM   30000
#define M_DIM   (B_DIM * L_DIM)   // 2000

__device__ __forceinline__ float sigmoidf(float x) { return 1.0f / (1.0f + __expf(-x)); }

// ---------------------------------------------------------------------------
// Kernel 1: main LSTM over (B=100, L=20, E=100). One block per batch row.
// ---------------------------------------------------------------------------
__global__ void lstm_main(const int* __restrict__ sess,
                          const float* __restrict__ item_emb,
                          const float* __restrict__ W_ih,
                          const float* __restrict__ W_hh,
                          const float* __restrict__ b_ih,
                          const float* __restrict__ b_hh,
                          float* __restrict__ hu /* [M_DIM][E_DIM] */)
{
    __shared__ float xt[E_DIM], h[E_DIM], c[E_DIM], gate[NGATE];
    const int b = blockIdx.x;
    const int tid = threadIdx.x;

    for (int e = tid; e < E_DIM; e += blockDim.x) { h[e] = 0.0f; c[e] = 0.0f; }
    __syncthreads();

    for (int t = 0; t < L_DIM; ++t) {
        const int it = sess[b * L_DIM + t];
        for (int e = tid; e < E_DIM; e += blockDim.x) xt[e] = item_emb[it * E_DIM + e];
        __syncthreads();

        for (int j = tid; j < NGATE; j += blockDim.x) {
            float acc = b_ih[j] + b_hh[j];
            const float* wi = &W_ih[j * E_DIM];
            const float* wh = &W_hh[j * E_DIM];
            #pragma unroll 4
            for (int e = 0; e < E_DIM; ++e) acc += xt[e] * wi[e] + h[e] * wh[e];
            gate[j] = acc;
        }
        __syncthreads();

        for (int e = tid; e < E_DIM; e += blockDim.x) {
            const float gi = sigmoidf(gate[e]);
            const float gf = sigmoidf(gate[E_DIM + e]);
            const float gg = tanhf(gate[2 * E_DIM + e]);
            const float go = sigmoidf(gate[3 * E_DIM + e]);
            const float cn = gf * c[e] + gi * gg;
            const float hn = go * tanhf(cn);
            c[e] = cn;
            h[e] = hn;
            hu[(b * L_DIM + t) * E_DIM + e] = hn;
        }
        __syncthreads();
    }
}

// ---------------------------------------------------------------------------
// Kernel 2: layer-2 aggregation: one-step gate on x0 -> st2,
// then ls2 = relu(concat(lt2, st2) @ W1^T). One block per support row (500).
// ---------------------------------------------------------------------------
__global__ void agg_layer2(const int* __restrict__ sess2,
                           const int* __restrict__ nodes2,
                           const float* __restrict__ item_emb,
                           const float* __restrict__ user_emb,
                           const float* __restrict__ W_ih,
                           const float* __restrict__ b_ih,
                           const float* __restrict__ b_hh,
                           const float* __restrict__ W1,
                           float* __restrict__ ls2 /* [500][E_DIM] */)
{
    __shared__ float x0[E_DIM], cat[2 * E_DIM], gate[NGATE];
    const int r = blockIdx.x;
    const int tid = threadIdx.x;

    const int it = sess2[r * L_DIM + 0];
    const int u  = nodes2[r];
    for (int e = tid; e < E_DIM; e += blockDim.x) {
        x0[e]  = item_emb[it * E_DIM + e];
        cat[e] = user_emb[u * E_DIM + e];   // lt2
    }
    __syncthreads();

    for (int j = tid; j < NGATE; j += blockDim.x) {
        float acc = b_ih[j] + b_hh[j];
        const float* wi = &W_ih[j * E_DIM];
        #pragma unroll 4
        for (int e = 0; e < E_DIM; ++e) acc += x0[e] * wi[e];
        gate[j] = acc;
    }
    __syncthreads();

    for (int e = tid; e < E_DIM; e += blockDim.x) {
        const float gi = sigmoidf(gate[e]);
        const float gg = tanhf(gate[2 * E_DIM + e]);
        const float go = sigmoidf(gate[3 * E_DIM + e]);
        const float c0 = gi * gg;
        cat[E_DIM + e] = go * tanhf(c0);    // st2
    }
    __syncthreads();

    for (int j = tid; j < E_DIM; j += blockDim.x) {
        float acc = 0.0f;
        const float* w = &W1[j * 2 * E_DIM];
        #pragma unroll 4
        for (int t2 = 0; t2 < 2 * E_DIM; ++t2) acc += cat[t2] * w[t2];
        ls2[r * E_DIM + j] = fmaxf(acc, 0.0f);
    }
}

// ---------------------------------------------------------------------------
// Kernel 3: GAT over 6 vectors (5 neighbors from ls2 + self q = hu[b][0]).
// ---------------------------------------------------------------------------
__global__ void gat_kernel(const float* __restrict__ hu,
                           const float* __restrict__ ls2,
                           const float* __restrict__ Wg0,
                           float* __restrict__ h0 /* [B_DIM][E_DIM] */)
{
    __shared__ float ns[6 * E_DIM];
    __shared__ float p[8];
    __shared__ float ctx[E_DIM];
    const int n = blockIdx.x;
    const int tid = threadIdx.x;

    for (int e = tid; e < E_DIM; e += blockDim.x) {
        const float q = hu[(n * L_DIM + 0) * E_DIM + e];
        ns[5 * E_DIM + e] = q;
        #pragma unroll
        for (int k = 0; k < S2_DIM; ++k)
            ns[k * E_DIM + e] = ls2[(n * S2_DIM + k) * E_DIM + e];
    }
    __syncthreads();

    if (tid < 6) {
        float s = 0.0f;
        for (int e = 0; e < E_DIM; ++e) s += ns[5 * E_DIM + e] * ns[tid * E_DIM + e];
        p[tid] = s;
    }
    __syncthreads();

    if (tid == 0) {
        float mx = p[0];
        for (int k = 1; k < 6; ++k) mx = fmaxf(mx, p[k]);
        float sum = 0.0f;
        for (int k = 0; k < 6; ++k) { p[k] = __expf(p[k] - mx); sum += p[k]; }
        const float inv = 1.0f / sum;
        for (int k = 0; k < 6; ++k) p[k] *= inv;
    }
    __syncthreads();

    for (int e = tid; e < E_DIM; e += blockDim.x) {
        float s = 0.0f;
        #pragma unroll
        for (int k = 0; k < 6; ++k) s += p[k] * ns[k * E_DIM + e];
        ctx[e] = s;
    }
    __syncthreads();

    for (int j = tid; j < E_DIM; j += blockDim.x) {
        float s = 0.0f;
        for (int e = 0; e < E_DIM; ++e) s += ctx[e] * Wg0[e * E_DIM + j];
        h0[n * E_DIM + j] = fmaxf(s, 0.0f);
    }
}

// ---------------------------------------------------------------------------
// Kernel 4: sr[m][j] = concat(hu[m], h0[m/L]) @ W2^T. One block per row (2000).
// ---------------------------------------------------------------------------
__global__ void sr_kernel(const float* __restrict__ hu,
                          const float* __restrict__ h0,
                          const float* __restrict__ W2,
                          float* __restrict__ sr /* [M_DIM][E_DIM] */)
{
    __shared__ float cat[2 * E_DIM];
    const int m = blockIdx.x;
    const int tid = threadIdx.x;
    const int b = m / L_DIM;

    for (int e = tid; e < E_DIM; e += blockDim.x) {
        cat[e]         = hu[m * E_DIM + e];
        cat[E_DIM + e] = h0[b * E_DIM + e];
    }
    __syncthreads();

    for (int j = tid; j < E_DIM; j += blockDim.x) {
        float acc = 0.0f;
        const float* w = &W2[j * 2 * E_DIM];
        #pragma unroll 4
        for (int t2 = 0; t2 < 2 * E_DIM; ++t2) acc += cat[t2] * w[t2];
        sr[m * E_DIM + j] = acc;
    }
}

// ---------------------------------------------------------------------------
// Kernel 5: logits[m][v] = sum_e sr[m][e]*item_emb[v][e], times mask[m].
// M=2000, K=100, N=30000 fp32 GEMM via V_WMMA_F32_16X16X4_F32.
//
// Block = 5 waves (160 thr): 5 M-tiles (80 rows) x one 80-col N-strip.
// The 80x100 f32 B-tile (32 KB) is staged into LDS ONCE per block via the
// Tensor Data Mover (tensor_load_to_lds, TENSORcnt), then each wave runs
// 25 k-steps x 5 WMMAs reading B fragments with ds_load_b64.
//
// The TDM's LDS write is a pure hardware side effect, invisible to LLVM.
// Since Bs would otherwise have no stores in the module, LLVM folds all its
// loads to undef (whole-global reasoning; a "memory" asm clobber does NOT
// stop it because the non-escaping LDS global provably can't be named by the
// asm). We defeat it with a never-executed store behind a laundered guard.
// ---------------------------------------------------------------------------
#define NT 5                 // N-tiles per wave (16x80 strip)
#define MW 5                 // M-tiles (waves) per block
#define NSTRIP (16 * NT)     // 80 columns per block

#if defined(__has_builtin)
#if __has_builtin(__builtin_amdgcn_tensor_load_to_lds)
#define HAVE_TDM 1
#endif
#endif
#ifndef HAVE_TDM
#define HAVE_TDM 0
#endif

__global__ void __launch_bounds__(MW * 32)
logits_wmma(const float* __restrict__ sr,
            const float* __restrict__ item_emb,
            const int* __restrict__ mask,
            float* __restrict__ out)
{
    // Single shared allocation -> LDS byte offset 0 (used as D#.lds_addr).
    __shared__ float Bs[NSTRIP * E_DIM];   // [col][k], 32 KB

    const int lane  = threadIdx.x & 31;
    const int wave  = threadIdx.x >> 5;

    const int n0 = blockIdx.x * NSTRIP;                 // N-strip base (375 strips)
    const int m0 = (blockIdx.y * MW + wave) * 16;       // this wave's M-tile (125 total)

#if HAVE_TDM
    // Opaque "may store" guard: `never` is 0 at runtime, but after the asm the
    // compiler must treat both the guard and the index as unknown, so a
    // reachable store to Bs[unknown] exists => loads from Bs cannot be folded
    // to undef, and none can be proven no-alias => no value forwarding either.
    int never = 0;
    asm volatile("" : "+s"(never));
    if (never) Bs[never] = 0.0f;

    // ---- Stage B tile: copy item_emb rows n0..n0+79 (100 floats each) into Bs in LDS, row by row
    if (wave == 0) {
        const unsigned long long ga =
            (unsigned long long)(const void*)&item_emb[(size_t)n0 * E_DIM];
        u32x4 g0;
        g0.x = 1u;                                   // count=1, user mode, no gather
        g0.y = 0u;                                   // lds_addr = 0 (Bs at LDS base)
        g0.z = (unsigned)(ga & 0xFFFFFFFFull);       // global_addr[31:0]
        g0.w = (unsigned)((ga >> 32) & 0x01FFFFFFull) | (2u << 30); // [56:32] | type=2

        i32x8 g1;
        g1[0] = (2 << 16);                           // data_size = 4 bytes
        g1[1] = (E_DIM & 0xFFFF) << 16;              // tensor_dim0 = 100 (lo16)
        g1[2] = (V_DIM & 0xFFFF) << 16;              // tensor_dim0 hi=0 | tensor_dim1 lo16
        g1[3] = (E_DIM << 16);                       // tensor_dim1 hi=0 | tile_dim0 = 100
        g1[4] = NSTRIP;                              // tile_dim1 = 80 | tile_dim2 = 0
        g1[5] = E_DIM;                               // tensor_dim0_stride = 100 (lo32)
        g1[6] = 0;                                   // stride hi | tensor_dim1_stride lo
        g1[7] = 0;

        const i32x4 z4 = {};
#if __has_include(<hip/amd_detail/amd_gfx1250_TDM.h>)
        const i32x8 z8 = {};
        __builtin_amdgcn_tensor_load_to_lds(g0, g1, z4, z4, z8, 0);   // clang-23 form
#else
        __builtin_amdgcn_tensor_load_to_lds(g0, g1, z4, z4, 0);       // ROCm 7.2 form
#endif
        __builtin_amdgcn_s_wait_tensorcnt(0);
        asm volatile("" ::: "memory");
    }
#else
    // Fallback: cooperative vector staging (keeps kernel compiling w/o TDM).
    for (int i = threadIdx.x * 4; i < NSTRIP * E_DIM; i += (MW * 32) * 4) {
        const int y = i / E_DIM;
        const int x = i % E_DIM;
        Bs[i + 0] = item_emb[(size_t)(n0 + y) * E_DIM + x + 0];
        Bs[i + 1] = item_emb[(size_t)(n0 + y) * E_DIM + x + 1];
        Bs[i + 2] = item_emb[(size_t)(n0 + y) * E_DIM + x + 2];
        Bs[i + 3] = item_emb[(size_t)(n0 + y) * E_DIM + x + 3];
    }
#endif
    __syncthreads();
    asm volatile("" ::: "memory");

    // ---- WMMA main loop
    const int half   = (lane >> 4) & 1;      // 0: K=0,1  1: K=2,3
    const int lane16 = lane & 15;
    const int rA     = m0 + lane16;

    v8f acc[NT];
    const v8f vzero = {};
    #pragma unroll
    for (int t = 0; t < NT; ++t) acc[t] = vzero;

    const float* aRow = &sr[(size_t)rA * E_DIM];

    for (int k4 = 0; k4 < E_DIM; k4 += 4) {
        const int kk = k4 + (half << 1);
        v2f a;
        a.x = aRow[kk];
        a.y = aRow[kk + 1];
        #pragma unroll
        for (int t = 0; t < NT; ++t) {
            const float* bRow = &Bs[(t * 16 + lane16) * E_DIM];
            v2f bb;
            bb.x = bRow[kk];
            bb.y = bRow[kk + 1];
            acc[t] = __builtin_amdgcn_wmma_f32_16x16x4_f32(
                /*neg_a=*/false, a, /*neg_b=*/false, bb,
                /*c_mod=*/(short)0, acc[t],
                /*reuse_a=*/false, /*reuse_b=*/false);
        }
    }

    // ---- Store (mask is all-ones int, applied per output row)
    const int colBase = n0 + lane16;
    #pragma unroll
    for (int t = 0; t < NT; ++t) {
        #pragma unroll
        for (int r = 0; r < 8; ++r) {
            const int row = m0 + r + half * 8;
            const float mk = (float)mask[row];
            out[(size_t)row * V_DIM + colBase + t * 16] = acc[t][r] * mk;
        }
    }
}

// ---------------------------------------------------------------------------
// Launch
// ---------------------------------------------------------------------------
extern "C" void kernel_launch(void* const* d_in, const int* in_sizes, int n_in,
                              void* d_out, int out_size, void* d_ws, size_t ws_size,
                              hipStream_t stream) {
    const int*   input_session = (const int*)d_in[0];
    // d_in[1], d_in[3]: layer-1 supports -- unused by the reference output
    const int*   nodes2        = (const int*)d_in[2];
    const int*   sess2         = (const int*)d_in[4];
    const int*   mask_y        = (const int*)d_in[5];
    const float* user_emb      = (const float*)d_in[6];
    const float* item_emb      = (const float*)d_in[7];
    const float* W_ih          = (const float*)d_in[8];
    const float* W_hh          = (const float*)d_in[9];
    const float* b_ih          = (const float*)d_in[10];
    const float* b_hh          = (const float*)d_in[11];
    const float* W1            = (const float*)d_in[12];
    const float* W2            = (const float*)d_in[13];
    const float* Wg0           = (const float*)d_in[14];
    float*       out           = (float*)d_out;

    // Workspace layout (bytes):
    char* ws = (char*)d_ws;
    float* hu  = (float*)(ws);               // 2000*100*4 = 800000
    float* ls2 = (float*)(ws + 800000);      // 500*100*4  = 200000
    float* h0  = (float*)(ws + 1000000);     // 100*100*4  = 40000
    float* sr  = (float*)(ws + 1040000);     // 2000*100*4 = 800000

    lstm_main<<<B_DIM, 128, 0, stream>>>(input_session, item_emb, W_ih, W_hh, b_ih, b_hh, hu);
    agg_layer2<<<B_DIM * S2_DIM, 128, 0, stream>>>(sess2, nodes2, item_emb, user_emb,
                                                   W_ih, b_ih, b_hh, W1, ls2);
    gat_kernel<<<B_DIM, 128, 0, stream>>>(hu, ls2, Wg0, h0);
    sr_kernel<<<M_DIM, 128, 0, stream>>>(hu, h0, W2, sr);

    // 375 N-strips x 25 M-supertiles; 5 waves/block, one M-tile per wave.
    dim3 grid(V_DIM / NSTRIP, (M_DIM / 16) / MW);    // (375, 25)
    logits_wmma<<<grid, MW * 32, 0, stream>>>(sr, item_emb, mask_y, out);
}